// VariationalQuantumCircuit_42511586295896
// MI455X (gfx1250) — compile-verified
//
#include <hip/hip_runtime.h>
#include <cstdint>

#define NQ    22
#define NST   (1u << NQ)          // 4,194,304 amplitudes
#define CB    13                  // chunk bits
#define CH    (1u << CB)          // 8192 amplitudes per LDS chunk (64 KB re+im)
#define PAIRS (CH / 2)
#define NTHR  256
#define NBLK  (NST / CH)          // 512 chunks

typedef unsigned int v4u __attribute__((ext_vector_type(4)));
typedef int          v4i __attribute__((ext_vector_type(4)));
typedef int          v8i __attribute__((ext_vector_type(8)));
typedef float        v2f __attribute__((ext_vector_type(2)));
typedef float        v8f __attribute__((ext_vector_type(8)));

#ifndef __has_builtin
#define __has_builtin(x) 0
#endif

#define HAVE_TDM (__has_builtin(__builtin_amdgcn_tensor_load_to_lds) && \
                  __has_builtin(__builtin_amdgcn_tensor_store_from_lds))
#define HAVE_WMMA __has_builtin(__builtin_amdgcn_wmma_f32_16x16x4_f32)

__device__ __forceinline__ void tdm_wait_all() {
#if __has_builtin(__builtin_amdgcn_s_wait_tensorcnt)
  __builtin_amdgcn_s_wait_tensorcnt((short)0);
#else
  asm volatile("s_wait_tensorcnt 0x0" ::: "memory");
#endif
}

#if HAVE_TDM
// One TDM descriptor: move a (tile_w x rows) tile of 4-byte elements between
// global memory (row stride = stride_e elements) and a contiguous LDS region.
__device__ __forceinline__ void tdm_copy(bool to_lds, unsigned lds_byte,
                                         const void* gp, unsigned tile_w,
                                         unsigned rows, unsigned stride_e) {
  unsigned long long ga = (unsigned long long)gp;
  v4u g0;
  g0[0] = 1u;                                            // count=1, user descriptor
  g0[1] = lds_byte;                                      // lds_addr
  g0[2] = (unsigned)ga;                                  // global_addr[31:0]
  g0[3] = ((unsigned)(ga >> 32) & 0x01FFFFFFu) | 0x80000000u; // addr[56:32] | type=2
  v8i g1;
  g1[0] = (int)(2u << 16);                               // wg_mask=0, data_size=4B
  g1[1] = (int)((tile_w & 0xFFFFu) << 16);               // tensor_dim0[15:0]
  g1[2] = (int)((tile_w >> 16) | ((rows & 0xFFFFu) << 16)); // td0 hi | tensor_dim1 lo
  g1[3] = (int)((rows >> 16) | ((tile_w & 0xFFFFu) << 16)); // td1 hi | tile_dim0
  g1[4] = (int)(rows & 0xFFFFu);                         // tile_dim1 (tile_dim2=0)
  g1[5] = (int)stride_e;                                 // tensor_dim0_stride lo
  g1[6] = 0;
  g1[7] = 0;
  v4i z4 = {0, 0, 0, 0};
#if defined(__clang_major__) && (__clang_major__ >= 23)
  v8i z8 = {0, 0, 0, 0, 0, 0, 0, 0};
  if (to_lds) __builtin_amdgcn_tensor_load_to_lds(g0, g1, z4, z4, z8, 0);
  else        __builtin_amdgcn_tensor_store_from_lds(g0, g1, z4, z4, z8, 0);
#else
  if (to_lds) __builtin_amdgcn_tensor_load_to_lds(g0, g1, z4, z4, 0);
  else        __builtin_amdgcn_tensor_store_from_lds(g0, g1, z4, z4, 0);
#endif
}
#endif

// Alternating-parity role mask for the P^-2-conjugated layer-2 gates:
// parity of x_j for j<=q, j == q (mod 2).
__device__ __forceinline__ unsigned rsmask(int q) {
  unsigned alt = (q & 1) ? 0xAAAAAAAAu : 0x55555555u;
  return alt & ((2u << q) - 1u);
}

// Butterfly over LDS chunk, pass-1 flavor (global role bits == local bits).
// Pair (i, i^mask), representative i has bit pb == 0:
//   a' = c*a - sigma*s*b ;  b' = c*b + sigma*s*a ,  sigma = (-1)^parity(i & rmask)
__device__ __forceinline__ void bfly1(float* lre, float* lim, int tid,
                                      unsigned mask, int pb, unsigned rmask,
                                      float c, float s) {
#pragma unroll 4
  for (int it = 0; it < (int)(PAIRS / NTHR); ++it) {
    unsigned t = (unsigned)(tid + it * NTHR);
    unsigned i = ((t >> pb) << (pb + 1)) | (t & ((1u << pb) - 1u));
    unsigned j = i ^ mask;
    float sg = (__popc(i & rmask) & 1) ? -s : s;
    float ar = lre[i], ai = lim[i], br = lre[j], bi = lim[j];
    lre[i] = c * ar - sg * br;
    lim[i] = c * ai - sg * bi;
    lre[j] = c * br + sg * ar;
    lim[j] = c * bi + sg * ai;
  }
  __syncthreads();
}

// Pass-2 flavor: local index l maps to global x = (l&3) | (blk<<2) | ((l>>2)<<11);
// role parity uses the full global index.
__device__ __forceinline__ void bfly2(float* lre, float* lim, int tid, unsigned xb,
                                      unsigned lmask, int pb, unsigned rmask,
                                      float c, float s) {
#pragma unroll 4
  for (int it = 0; it < (int)(PAIRS / NTHR); ++it) {
    unsigned t = (unsigned)(tid + it * NTHR);
    unsigned i = ((t >> pb) << (pb + 1)) | (t & ((1u << pb) - 1u));
    unsigned j = i ^ lmask;
    unsigned x = (i & 3u) | xb | ((i >> 2) << 11);
    float sg = (__popc(x & rmask) & 1) ? -s : s;
    float ar = lre[i], ai = lim[i], br = lre[j], bi = lim[j];
    lre[i] = c * ar - sg * br;
    lim[i] = c * ai - sg * bi;
    lre[j] = c * br + sg * ar;
    lim[j] = c * bi + sg * ai;
  }
  __syncthreads();
}

#if HAVE_WMMA
// Apply fused 16x16 matrix M (row-major in Mg) along bits 0..3 of a plane:
// for every 16-float column v (one per rest-index), v <- M v, via
// D = sum_kb A_kb(16x4) x B_kb(4x16) chained on V_WMMA_F32_16X16X4_F32.
// Each wave owns tiles t = wid, wid+8, ... (16 columns per tile): in-place safe.
__device__ __forceinline__ void apply_m16(float* plane, const float* __restrict__ Mg,
                                          int lane, int wid) {
  const int row  = lane & 15;
  const int koff = (lane >> 4) << 1;   // lanes 16..31 carry K = 2,3 of each block
  v2f a0, a1, a2, a3;
  a0[0] = Mg[row * 16 + 0  + koff]; a0[1] = Mg[row * 16 + 1  + koff];
  a1[0] = Mg[row * 16 + 4  + koff]; a1[1] = Mg[row * 16 + 5  + koff];
  a2[0] = Mg[row * 16 + 8  + koff]; a2[1] = Mg[row * 16 + 9  + koff];
  a3[0] = Mg[row * 16 + 12 + koff]; a3[1] = Mg[row * 16 + 13 + koff];
  const int n  = lane & 15;
  const int gh = (lane >> 4) << 3;     // D rows 0..7 (lanes<16) or 8..15
  for (int t = wid; t < (int)(CH / 16 / 16); t += 8) {
    float* colp = plane + (t * 16 + n) * 16;
    v2f b0 = {colp[0  + koff], colp[1  + koff]};
    v2f b1 = {colp[4  + koff], colp[5  + koff]};
    v2f b2 = {colp[8  + koff], colp[9  + koff]};
    v2f b3 = {colp[12 + koff], colp[13 + koff]};
    v8f acc = {};
    acc = __builtin_amdgcn_wmma_f32_16x16x4_f32(false, a0, false, b0, (short)0, acc, false, false);
    acc = __builtin_amdgcn_wmma_f32_16x16x4_f32(false, a1, false, b1, (short)0, acc, false, false);
    acc = __builtin_amdgcn_wmma_f32_16x16x4_f32(false, a2, false, b2, (short)0, acc, false, false);
    acc = __builtin_amdgcn_wmma_f32_16x16x4_f32(false, a3, false, b3, (short)0, acc, false, false);
    float* outp = colp + gh;
    *(float4*)(outp)     = make_float4(acc[0], acc[1], acc[2], acc[3]);
    *(float4*)(outp + 4) = make_float4(acc[4], acc[5], acc[6], acc[7]);
  }
}
#endif

// ---- Setup: cos/sin table + fused 16x16 matrix for the 9 low-bit gates ------
__global__ __launch_bounds__(256) void vqc_setup(const float* __restrict__ prm,
                                                 float* __restrict__ Mg,
                                                 float* __restrict__ gcs) {
  __shared__ float Msh[256];
  __shared__ float c66[66], s66[66];
  const int tid = threadIdx.x;
  if (tid < 66) {
    float a = 0.5f * prm[tid];
    float c = cosf(a), s = sinf(a);
    c66[tid] = c; s66[tid] = s;
    gcs[tid] = c; gcs[66 + tid] = s;
  }
  Msh[tid] = ((tid >> 4) == (tid & 15)) ? 1.0f : 0.0f;   // identity
  __syncthreads();
  // Chronological: L0 q0..3, L1' q0..2, L2'' q0..1  (M <- G * M each step)
  const int qs[9]   = {0, 1, 2, 3, 0, 1, 2, 0, 1};
  const int kind[9] = {0, 0, 0, 0, 1, 1, 1, 2, 2};
  for (int g = 0; g < 9; ++g) {
    int q = qs[g];
    unsigned mask, rmask; int pidx;
    if (kind[g] == 0)      { mask = 1u << q; rmask = 0u;             pidx = q; }
    else if (kind[g] == 1) { mask = 3u << q; rmask = (2u << q) - 1u; pidx = 22 + q; }
    else                   { mask = 5u << q; rmask = rsmask(q);      pidx = 44 + q; }
    float c = c66[pidx], s = s66[pidx];
    if (tid < 128) {
      unsigned p = (unsigned)(tid >> 4), col = (unsigned)(tid & 15);
      unsigned i = ((p >> q) << (q + 1)) | (p & ((1u << q) - 1u));
      unsigned j = i ^ mask;
      float sg = (__popc(i & rmask) & 1) ? -s : s;
      float a0 = Msh[i * 16 + col], b0 = Msh[j * 16 + col];
      Msh[i * 16 + col] = c * a0 - sg * b0;
      Msh[j * 16 + col] = c * b0 + sg * a0;
    }
    __syncthreads();
  }
  Mg[tid] = Msh[tid];
}

// ---------------- Pass 1: gates whose pair-masks live in bits 0..12 ----------
__global__ __launch_bounds__(NTHR) void vqc_pass1(const float* __restrict__ sre,
                                                  const float* __restrict__ sim,
                                                  const float* __restrict__ gcs,
                                                  const float* __restrict__ Mg,
                                                  float* __restrict__ wre,
                                                  float* __restrict__ wim) {
  __shared__ float lre[CH];
  __shared__ float lim[CH];
  __shared__ float gc[66], gs[66];
  const int tid = threadIdx.x;
  const size_t base = (size_t)blockIdx.x << CB;

#if HAVE_TDM
  unsigned wid = (unsigned)__builtin_amdgcn_readfirstlane((int)(threadIdx.x >> 5));
  unsigned ldsre = ((unsigned)(unsigned long long)(void*)lre) + wid * 4096u;
  unsigned ldsim = ((unsigned)(unsigned long long)(void*)lim) + wid * 4096u;
  tdm_copy(true, ldsre, sre + base + (size_t)wid * 1024u, 1024u, 1u, 1024u);
  tdm_copy(true, ldsim, sim + base + (size_t)wid * 1024u, 1024u, 1u, 1024u);
#else
  unsigned wid = (unsigned)(threadIdx.x >> 5);
  for (int k = tid; k < (int)(CH / 4); k += NTHR) {
    ((float4*)lre)[k] = ((const float4*)(sre + base))[k];
    ((float4*)lim)[k] = ((const float4*)(sim + base))[k];
  }
#endif
  if (tid < 66) { gc[tid] = gcs[tid]; gs[tid] = gcs[66 + tid]; }
#if HAVE_TDM
  tdm_wait_all();
#endif
  __syncthreads();

#if HAVE_WMMA
  // Fused 16x16 (L0 q0-3, L1' q0-2, L2'' q0-1) on bits 0..3 via WMMA.
  {
    const int lane = tid & 31;
    apply_m16(lre, Mg, lane, (int)wid);
    apply_m16(lim, Mg, lane, (int)wid);
  }
  __syncthreads();
  const int q0a = 4, q0b = 3, q0c = 2;
#else
  (void)Mg;
  const int q0a = 0, q0b = 0, q0c = 0;
#endif

  // Layer 0: plain RY (mask e_q, representative is role 0)
  for (int q = q0a; q <= 12; ++q)
    bfly1(lre, lim, tid, 1u << q, q, 0u, gc[q], gs[q]);
  // Layer 1 conjugated by P^-1: mask {q,q+1}, role = parity(x[0..q])
  for (int q = q0b; q <= 11; ++q)
    bfly1(lre, lim, tid, 3u << q, q, (2u << q) - 1u, gc[22 + q], gs[22 + q]);
  // Layer 2 conjugated by P^-2: mask {q,q+2}, role = same-parity prefix parity
  for (int q = q0c; q <= 10; ++q)
    bfly1(lre, lim, tid, 5u << q, q, rsmask(q), gc[44 + q], gs[44 + q]);

#if HAVE_TDM
  tdm_copy(false, ldsre, wre + base + (size_t)wid * 1024u, 1024u, 1u, 1024u);
  tdm_copy(false, ldsim, wim + base + (size_t)wid * 1024u, 1024u, 1u, 1024u);
  tdm_wait_all();
#else
  for (int k = tid; k < (int)(CH / 4); k += NTHR) {
    ((float4*)(wre + base))[k] = ((float4*)lre)[k];
    ((float4*)(wim + base))[k] = ((float4*)lim)[k];
  }
#endif
}

// ------- Pass 2: local bits {0,1} U {11..21}; all remaining pair-masks local --
__global__ __launch_bounds__(NTHR) void vqc_pass2(const float* __restrict__ gcs,
                                                  float* __restrict__ wre,
                                                  float* __restrict__ wim) {
  __shared__ float lre[CH];
  __shared__ float lim[CH];
  __shared__ float gc[66], gs[66];
  const int tid = threadIdx.x;
  const unsigned xb = (unsigned)blockIdx.x << 2;  // global bits 2..10

#if HAVE_TDM
  unsigned wid = (unsigned)__builtin_amdgcn_readfirstlane((int)(threadIdx.x >> 5));
  unsigned ldsre = ((unsigned)(unsigned long long)(void*)lre) + wid * 4096u;
  unsigned ldsim = ((unsigned)(unsigned long long)(void*)lim) + wid * 4096u;
  // Wave slice: 256 rows of 4 elements, row stride 2^11 elements.
  size_t ebase = (size_t)xb + ((size_t)wid << 19);
  tdm_copy(true, ldsre, wre + ebase, 4u, 256u, 2048u);
  tdm_copy(true, ldsim, wim + ebase, 4u, 256u, 2048u);
#else
  for (int l = tid; l < (int)CH; l += NTHR) {
    unsigned x = ((unsigned)l & 3u) | xb | (((unsigned)l >> 2) << 11);
    lre[l] = wre[x];
    lim[l] = wim[x];
  }
#endif
  if (tid < 66) { gc[tid] = gcs[tid]; gs[tid] = gcs[66 + tid]; }
#if HAVE_TDM
  tdm_wait_all();
#endif
  __syncthreads();

  // Layer 0: qubits 13..21 (global bit b -> local bit b-9)
  for (int q = 13; q <= 21; ++q) {
    unsigned xm = 1u << q;
    bfly2(lre, lim, tid, xb, xm >> 9, q - 9, 0u, gc[q], gs[q]);
  }
  // Layer 1': q = 12..21, mask {q,q+1} (clipped to 22 bits)
  for (int q = 12; q <= 21; ++q) {
    unsigned xm = (3u << q) & (NST - 1u);
    bfly2(lre, lim, tid, xb, xm >> 9, q - 9, (2u << q) - 1u, gc[22 + q], gs[22 + q]);
  }
  // Layer 2'': q = 11..21, mask {q,q+2} (clipped)
  for (int q = 11; q <= 21; ++q) {
    unsigned xm = (5u << q) & (NST - 1u);
    bfly2(lre, lim, tid, xb, xm >> 9, q - 9, rsmask(q), gc[44 + q], gs[44 + q]);
  }

#if HAVE_TDM
  tdm_copy(false, ldsre, wre + ebase, 4u, 256u, 2048u);
  tdm_copy(false, ldsim, wim + ebase, 4u, 256u, 2048u);
  tdm_wait_all();
#else
  for (int l = tid; l < (int)CH; l += NTHR) {
    unsigned x = ((unsigned)l & 3u) | xb | (((unsigned)l >> 2) << 11);
    wre[x] = lre[l];
    wim[x] = lim[l];
  }
#endif
}

// ---- Pass 3: apply trailing P^2 (out[z] = u[z ^ (z<<2)]) and interleave -----
__global__ __launch_bounds__(256) void vqc_pass3(const float* __restrict__ wre,
                                                 const float* __restrict__ wim,
                                                 float2* __restrict__ out) {
  unsigned z = blockIdx.x * 256u + threadIdx.x;
  unsigned x = (z ^ (z << 2)) & (NST - 1u);
  out[z] = make_float2(wre[x], wim[x]);
}

extern "C" void kernel_launch(void* const* d_in, const int* in_sizes, int n_in,
                              void* d_out, int out_size, void* d_ws, size_t ws_size,
                              hipStream_t stream) {
  (void)in_sizes; (void)n_in; (void)out_size; (void)ws_size;
  const float* prm = (const float*)d_in[0];   // 66 angles
  const float* sre = (const float*)d_in[1];   // 2^22 real
  const float* sim = (const float*)d_in[2];   // 2^22 imag
  float* wre = (float*)d_ws;                  // 16 MB plane
  float* wim = wre + NST;                     // 16 MB plane
  float* Mg  = wim + NST;                     // 256 floats: fused 16x16 matrix
  float* gcs = Mg + 256;                      // 132 floats: cos/sin table

  vqc_setup<<<1, 256, 0, stream>>>(prm, Mg, gcs);
  vqc_pass1<<<NBLK, NTHR, 0, stream>>>(sre, sim, gcs, Mg, wre, wim);
  vqc_pass2<<<NBLK, NTHR, 0, stream>>>(gcs, wre, wim);
  vqc_pass3<<<NST / 256, 256, 0, stream>>>(wre, wim, (float2*)d_out);
}